// SAGE_34892314312968
// MI455X (gfx1250) — compile-verified
//
#include <hip/hip_runtime.h>
#include <hip/hip_bf16.h>

typedef __attribute__((ext_vector_type(2))) float v2f;
typedef __attribute__((ext_vector_type(8))) float v8f;

#define IN_C 128

// ---------------------------------------------------------------------------
// Kernel 1: embedding-bag sum (16 words) + LayerNorm + ReLU.
// One wave32 per output row; lane l owns channels [4l, 4l+4) as float4.
// ---------------------------------------------------------------------------
__global__ void k_embed_ln(const int* __restrict__ x, const float* __restrict__ emb,
                           const float* __restrict__ g, const float* __restrict__ b,
                           float* __restrict__ h0, int n0) {
  int wave = (int)((blockIdx.x * blockDim.x + threadIdx.x) >> 5);
  int lane = threadIdx.x & 31;
  if (wave >= n0) return;

  const int* xr = x + (size_t)wave * 16;
  float4 acc = make_float4(0.f, 0.f, 0.f, 0.f);
#pragma unroll
  for (int w = 0; w < 16; ++w) {
    int idx = xr[w];  // uniform per wave -> scalar load
    float4 v = ((const float4*)(emb + (size_t)idx * IN_C))[lane];
    acc.x += v.x; acc.y += v.y; acc.z += v.z; acc.w += v.w;
  }
  float s  = acc.x + acc.y + acc.z + acc.w;
  float ss = acc.x * acc.x + acc.y * acc.y + acc.z * acc.z + acc.w * acc.w;
#pragma unroll
  for (int off = 16; off > 0; off >>= 1) {
    s  += __shfl_xor(s,  off, 32);
    ss += __shfl_xor(ss, off, 32);
  }
  float mu  = s * (1.0f / IN_C);
  float var = ss * (1.0f / IN_C) - mu * mu;
  float inv = rsqrtf(var + 1e-5f);

  float4 gg = ((const float4*)g)[lane];
  float4 bb = ((const float4*)b)[lane];
  float4 o;
  o.x = fmaxf((acc.x - mu) * inv * gg.x + bb.x, 0.f);
  o.y = fmaxf((acc.y - mu) * inv * gg.y + bb.y, 0.f);
  o.z = fmaxf((acc.z - mu) * inv * gg.z + bb.z, 0.f);
  o.w = fmaxf((acc.w - mu) * inv * gg.w + bb.w, 0.f);
  ((float4*)(h0 + (size_t)wave * IN_C))[lane] = o;
}

// ---------------------------------------------------------------------------
// Kernel 2: edge scatter-add (segment_sum of messages + degree count).
// One wave32 per edge; float atomics into agg/cnt.
// ---------------------------------------------------------------------------
__global__ void k_scatter(const int* __restrict__ src, const int* __restrict__ dst,
                          const float* __restrict__ h, float* __restrict__ agg,
                          float* __restrict__ cnt, int ne) {
  int e = (int)((blockIdx.x * blockDim.x + threadIdx.x) >> 5);
  int lane = threadIdx.x & 31;
  if (e >= ne) return;
  int sIdx = src[e];
  int dIdx = dst[e];
  float4 v = ((const float4*)(h + (size_t)sIdx * IN_C))[lane];
  float* ap = agg + (size_t)dIdx * IN_C + lane * 4;
  atomicAdd(ap + 0, v.x);
  atomicAdd(ap + 1, v.y);
  atomicAdd(ap + 2, v.z);
  atomicAdd(ap + 3, v.w);
  if (lane == 0) atomicAdd(cnt + dIdx, 1.0f);
}

// ---------------------------------------------------------------------------
// Kernel 3: SAGE linear stage with fp32 WMMA:
//   out[i,:] = (agg[i,:]/max(cnt[i],1)) @ Wl.T + hself[i,:] @ Wr.T + bl   (+ReLU)
// 2 waves/block, one 16x128 output strip per wave.
// K dimension processed in chunks of 16 fragments: each chunk batch-loads 16
// weight fragments into a FULLY-UNROLLED register array (constant indices ->
// guaranteed SROA to VGPRs, no scratch), then WMMAs against LDS activations.
// ---------------------------------------------------------------------------
#define LDS_STRIDE 132

__global__ void __launch_bounds__(64, 1)
k_sage_gemm(const float* __restrict__ agg, const float* __restrict__ cnt,
            const float* __restrict__ hself,
            const float* __restrict__ Wl, const float* __restrict__ bl,
            const float* __restrict__ Wr,
            float* __restrict__ out, int n_tgt, int do_relu) {
  __shared__ float lmean[2][16][LDS_STRIDE];
  __shared__ float lh[2][16][LDS_STRIDE];

  int wid  = threadIdx.x >> 5;           // wave in block: 0..1
  int lane = threadIdx.x & 31;
  int tile = blockIdx.x * 2 + wid;       // 16-row output tile
  int row0 = tile * 16;
  if (row0 >= n_tgt) return;             // wave-uniform exit (no barriers used)

  // Stage mean and self tiles into LDS (clamp rows for the ragged last tile).
#pragma unroll 4
  for (int r = 0; r < 16; ++r) {
    int row = row0 + r;
    if (row >= n_tgt) row = n_tgt - 1;
    float inv = 1.0f / fmaxf(cnt[row], 1.0f);
    float4 a  = ((const float4*)(agg   + (size_t)row * IN_C))[lane];
    float4 hh = ((const float4*)(hself + (size_t)row * IN_C))[lane];
    float* mrow = &lmean[wid][r][0];
    float* hr   = &lh[wid][r][0];
    mrow[lane * 4 + 0] = a.x * inv;
    mrow[lane * 4 + 1] = a.y * inv;
    mrow[lane * 4 + 2] = a.z * inv;
    mrow[lane * 4 + 3] = a.w * inv;
    hr[lane * 4 + 0] = hh.x;
    hr[lane * 4 + 1] = hh.y;
    hr[lane * 4 + 2] = hh.z;
    hr[lane * 4 + 3] = hh.w;
  }
  // Same-wave LDS RAW: DS ops from one wave are processed in order.

  int m  = lane & 15;          // output column within 16-wide N tile / A row
  int hi = (lane >> 4) & 1;    // lane-half selector per WMMA fp32 layout

  const float* amean = &lmean[wid][m][0];
  const float* aself = &lh[wid][m][0];

  for (int nt = 0; nt < 8; ++nt) {
    int n = nt * 16 + m;
    const float* wlrow = Wl + (size_t)n * IN_C;   // W[n][k], contiguous in k
    const float* wrrow = Wr + (size_t)n * IN_C;

    v8f c = {};

    // ---- pass 1: mean @ Wl.T, two chunks of 16 K-fragments ----
#pragma unroll
    for (int ch = 0; ch < 2; ++ch) {
      v2f bw[16];                        // constant-indexed -> VGPRs
#pragma unroll
      for (int j = 0; j < 16; ++j)       // 16 independent global_load_b64
        bw[j] = *(const v2f*)(wlrow + (ch * 16 + j) * 4 + hi * 2);
#pragma unroll
      for (int j = 0; j < 16; ++j) {
        v2f a = *(const v2f*)(amean + (ch * 16 + j) * 4 + hi * 2); // ds_load_b64
        c = __builtin_amdgcn_wmma_f32_16x16x4_f32(false, a, false, bw[j],
                                                  (short)0, c, false, false);
      }
    }
    // ---- pass 2: hself @ Wr.T ----
#pragma unroll
    for (int ch = 0; ch < 2; ++ch) {
      v2f bw[16];
#pragma unroll
      for (int j = 0; j < 16; ++j)
        bw[j] = *(const v2f*)(wrrow + (ch * 16 + j) * 4 + hi * 2);
#pragma unroll
      for (int j = 0; j < 16; ++j) {
        v2f a = *(const v2f*)(aself + (ch * 16 + j) * 4 + hi * 2);
        c = __builtin_amdgcn_wmma_f32_16x16x4_f32(false, a, false, bw[j],
                                                  (short)0, c, false, false);
      }
    }

    float bias = bl[n];
#pragma unroll
    for (int r = 0; r < 8; ++r) {
      int row = row0 + r + hi * 8;      // C layout: M = r + 8*(lane>=16)
      if (row < n_tgt) {
        float val = c[r] + bias;
        if (do_relu) val = fmaxf(val, 0.f);
        out[(size_t)row * IN_C + n] = val;
      }
    }
  }
}

// ---------------------------------------------------------------------------
// Host-side launcher
// ---------------------------------------------------------------------------
extern "C" void kernel_launch(void* const* d_in, const int* in_sizes, int n_in,
                              void* d_out, int out_size, void* d_ws, size_t ws_size,
                              hipStream_t stream) {
  // Input order: x, src1, dst1, src2, dst2, n1, n2, emb, ln_g, ln_b,
  //              Wl1, bl1, Wr1, Wl2, bl2, Wr2
  const int*   x    = (const int*)d_in[0];
  const int*   src1 = (const int*)d_in[1];
  const int*   dst1 = (const int*)d_in[2];
  const int*   src2 = (const int*)d_in[3];
  const int*   dst2 = (const int*)d_in[4];
  const float* emb  = (const float*)d_in[7];
  const float* ln_g = (const float*)d_in[8];
  const float* ln_b = (const float*)d_in[9];
  const float* Wl1  = (const float*)d_in[10];
  const float* bl1  = (const float*)d_in[11];
  const float* Wr1  = (const float*)d_in[12];
  const float* Wl2  = (const float*)d_in[13];
  const float* bl2  = (const float*)d_in[14];
  const float* Wr2  = (const float*)d_in[15];
  float* out = (float*)d_out;

  const int n0 = in_sizes[0] / 16;   // 100000
  const int e1 = in_sizes[1];        // 500000
  const int e2 = in_sizes[3];        // 100000
  const int n1 = 25000;              // module constants in the reference
  const int n2 = 5000;

  // Workspace carve-up (256B aligned slices).
  char* ws = (char*)d_ws;
  auto align256 = [](size_t v) { return (v + 255) & ~(size_t)255; };
  size_t off = 0;
  float* h0   = (float*)(ws + off); off += align256((size_t)n0 * IN_C * 4);
  float* agg1 = (float*)(ws + off); off += align256((size_t)n1 * IN_C * 4);
  float* cnt1 = (float*)(ws + off); off += align256((size_t)n1 * 4);
  float* h1   = (float*)(ws + off); off += align256((size_t)n1 * IN_C * 4);
  float* agg2 = (float*)(ws + off); off += align256((size_t)n2 * IN_C * 4);
  float* cnt2 = (float*)(ws + off); off += align256((size_t)n2 * 4);
  (void)ws_size; (void)n_in; (void)out_size;

  // Zero accumulators every call (deterministic w.r.t. poisoned scratch).
  hipMemsetAsync(agg1, 0, (size_t)n1 * IN_C * 4, stream);
  hipMemsetAsync(cnt1, 0, (size_t)n1 * 4, stream);
  hipMemsetAsync(agg2, 0, (size_t)n2 * IN_C * 4, stream);
  hipMemsetAsync(cnt2, 0, (size_t)n2 * 4, stream);

  // 1) embedding bag + layernorm + relu -> h0 [n0 x 128]
  {
    dim3 blk(256);
    dim3 grd((unsigned)((n0 + 7) / 8));
    k_embed_ln<<<grd, blk, 0, stream>>>(x, emb, ln_g, ln_b, h0, n0);
  }
  // 2) conv1 scatter
  {
    dim3 blk(256);
    dim3 grd((unsigned)((e1 + 7) / 8));
    k_scatter<<<grd, blk, 0, stream>>>(src1, dst1, h0, agg1, cnt1, e1);
  }
  // 3) conv1 linear + relu -> h1 [n1 x 128]
  {
    int tiles = (n1 + 15) / 16;
    dim3 blk(64);
    dim3 grd((unsigned)((tiles + 1) / 2));
    k_sage_gemm<<<grd, blk, 0, stream>>>(agg1, cnt1, h0, Wl1, bl1, Wr1, h1, n1, 1);
  }
  // 4) conv2 scatter
  {
    dim3 blk(256);
    dim3 grd((unsigned)((e2 + 7) / 8));
    k_scatter<<<grd, blk, 0, stream>>>(src2, dst2, h1, agg2, cnt2, e2);
  }
  // 5) conv2 linear -> out [n2 x 128]
  {
    int tiles = (n2 + 15) / 16;
    dim3 blk(64);
    dim3 grd((unsigned)((tiles + 1) / 2));
    k_sage_gemm<<<grd, blk, 0, stream>>>(agg2, cnt2, h1, Wl2, bl2, Wr2, out, n2, 0);
  }
}